// RelGraphConvLayer_73504070304033
// MI455X (gfx1250) — compile-verified
//
#include <hip/hip_runtime.h>

// ---------------------------------------------------------------------------
// R-GCN layer for MAG-style heterograph on MI455X (gfx1250, wave32, WMMA).
//
// Per relation r:   proj = x_src @ W[r]            (fp32 WMMA 16x16x4, W in LDS)
//                   deg[d] = #edges into d          (atomic f32)
//                   invdeg[d] = 1/max(deg[d],1)     (one pass, <=200k elems)
//                   out[d] += proj[src] * invdeg[d] (float4 gather + f32 atomics,
//                                                    accumulators L2-resident)
// Then per type:    out = relu(out + x @ W_self + bias) (fused WMMA finalize)
// ---------------------------------------------------------------------------

typedef __attribute__((ext_vector_type(2))) float v2f;
typedef __attribute__((ext_vector_type(8))) float v8f;

#define K_DIM 128
#define N_DIM 64

// ---- shared GEMM core: acc[4] (16x64 tile) = X[rowBase:+16, :] @ W_lds -----
// W staged in LDS as pair layout: lw[kp][n][j] = W[2*kp+j][n], so each B
// fragment (two consecutive K values for one n) is one aligned ds_load_b64.
// WMMA f32 16x16x4 layouts (ISA 7.12.2):
//   A: lane L: m = L%16, holds K = {2*(L/16), 2*(L/16)+1} of the k-step
//   B: lane L: n = L%16, holds same two K values
//   D: lane L: n = L%16, VGPR v -> m = v + 8*(L/16)
__device__ __forceinline__ void stage_w_lds(const float* __restrict__ W,
                                            float (*lw)[N_DIM][2]) {
  for (int i = threadIdx.x; i < K_DIM * N_DIM / 4; i += 256) {
    const int base = i * 4;           // element index in row-major W[128][64]
    const int k = base >> 6;
    const int n0 = base & 63;
    const float4 v = *(const float4*)(W + base);
    lw[k >> 1][n0 + 0][k & 1] = v.x;
    lw[k >> 1][n0 + 1][k & 1] = v.y;
    lw[k >> 1][n0 + 2][k & 1] = v.z;
    lw[k >> 1][n0 + 3][k & 1] = v.w;
  }
}

__device__ __forceinline__ void gemm_tile(const float* __restrict__ xrow,
                                          const float (*lw)[N_DIM][2],
                                          int n, int hi,
                                          v8f& acc0, v8f& acc1, v8f& acc2, v8f& acc3) {
  const int koff = hi * 2;
  for (int k = 0; k < K_DIM; k += 4) {
    const v2f a = *(const v2f*)(xrow + k + koff);        // global_load_b64
    const v2f* wp = (const v2f*)&lw[(k >> 1) + hi][n][0]; // ds_load_b64 x4
    const v2f b0 = wp[0];
    const v2f b1 = wp[16];
    const v2f b2 = wp[32];
    const v2f b3 = wp[48];
    acc0 = __builtin_amdgcn_wmma_f32_16x16x4_f32(false, a, false, b0, (short)0, acc0, false, false);
    acc1 = __builtin_amdgcn_wmma_f32_16x16x4_f32(false, a, false, b1, (short)0, acc1, false, false);
    acc2 = __builtin_amdgcn_wmma_f32_16x16x4_f32(false, a, false, b2, (short)0, acc2, false, false);
    acc3 = __builtin_amdgcn_wmma_f32_16x16x4_f32(false, a, false, b3, (short)0, acc3, false, false);
  }
}

// --------------------------- WMMA GEMM: Y = X @ W ---------------------------
__global__ __launch_bounds__(256)
void gemm_proj_kernel(const float* __restrict__ X, const float* __restrict__ W,
                      float* __restrict__ Y, int M) {
  __shared__ float lw[K_DIM / 2][N_DIM][2];   // 32 KB
  stage_w_lds(W, lw);
  __syncthreads();                            // all waves reach this barrier

  const int lane = threadIdx.x & 31;
  const int wave = threadIdx.x >> 5;
  const int rowBase = (blockIdx.x * 8 + wave) * 16;
  if (rowBase >= M) return;                   // wave-uniform, after barrier
  const int n  = lane & 15;
  const int hi = lane >> 4;

  const float* xrow = X + (size_t)(rowBase + n) * K_DIM;
  v8f acc0 = {}, acc1 = {}, acc2 = {}, acc3 = {};
  gemm_tile(xrow, lw, n, hi, acc0, acc1, acc2, acc3);

  const int mBase = rowBase + hi * 8;
#pragma unroll
  for (int v = 0; v < 8; ++v) {
    float* yr = Y + (size_t)(mBase + v) * N_DIM + n;
    yr[0]  = acc0[v];
    yr[16] = acc1[v];
    yr[32] = acc2[v];
    yr[48] = acc3[v];
  }
}

// ------------------- finalize: out = relu(out + X@Ws + bias) ----------------
__global__ __launch_bounds__(256)
void finalize_kernel(const float* __restrict__ X, const float* __restrict__ W,
                     const float* __restrict__ bias, float* __restrict__ OUT,
                     int M) {
  __shared__ float lw[K_DIM / 2][N_DIM][2];
  stage_w_lds(W, lw);
  __syncthreads();

  const int lane = threadIdx.x & 31;
  const int wave = threadIdx.x >> 5;
  const int rowBase = (blockIdx.x * 8 + wave) * 16;
  if (rowBase >= M) return;
  const int n  = lane & 15;
  const int hi = lane >> 4;

  const float* xrow = X + (size_t)(rowBase + n) * K_DIM;
  v8f acc0 = {}, acc1 = {}, acc2 = {}, acc3 = {};
  gemm_tile(xrow, lw, n, hi, acc0, acc1, acc2, acc3);

  const float bs0 = bias[n];
  const float bs1 = bias[16 + n];
  const float bs2 = bias[32 + n];
  const float bs3 = bias[48 + n];
  const int mBase = rowBase + hi * 8;
#pragma unroll
  for (int v = 0; v < 8; ++v) {
    float* yr = OUT + (size_t)(mBase + v) * N_DIM + n;
    yr[0]  = fmaxf(yr[0]  + acc0[v] + bs0, 0.0f);
    yr[16] = fmaxf(yr[16] + acc1[v] + bs1, 0.0f);
    yr[32] = fmaxf(yr[32] + acc2[v] + bs2, 0.0f);
    yr[48] = fmaxf(yr[48] + acc3[v] + bs3, 0.0f);
  }
}

// ----------------------------- degree count ---------------------------------
__global__ __launch_bounds__(256)
void degree_kernel(const int* __restrict__ dst, int ne, float* __restrict__ deg) {
  int e = blockIdx.x * 256 + threadIdx.x;
  if (e < ne) atomicAdd(&deg[dst[e]], 1.0f);
}

// --------------------- invdeg[i] = 1 / max(deg[i], 1) -----------------------
__global__ __launch_bounds__(256)
void invdeg_kernel(float* __restrict__ deg, int n) {
  int i = blockIdx.x * 256 + threadIdx.x;
  if (i < n) deg[i] = 1.0f / fmaxf(deg[i], 1.0f);
}

// ------------- scatter: out[dst] += proj[src] * invdeg[dst] -----------------
// 16 consecutive threads = one edge; float4 gather (global_load_b128) and
// 4 f32 atomics per thread. Destination accumulator regions (<= 51 MB) stay
// resident in the 192 MB L2, so atomics resolve on-chip.
__global__ __launch_bounds__(256)
void scatter_kernel(const float* __restrict__ proj, const int* __restrict__ src,
                    const int* __restrict__ dst, const float* __restrict__ invdeg,
                    float* __restrict__ out, int ne) {
  long tid = (long)blockIdx.x * 256 + threadIdx.x;
  int e = (int)(tid >> 4);
  int f = (int)(tid & 15) * 4;
  if (e >= ne) return;
  const int s = src[e];
  const int d = dst[e];
  const float w = invdeg[d];
  const float4 p = *(const float4*)(proj + (size_t)s * N_DIM + f);
  float* o = out + (size_t)d * N_DIM + f;
  atomicAdd(o + 0, p.x * w);
  atomicAdd(o + 1, p.y * w);
  atomicAdd(o + 2, p.z * w);
  atomicAdd(o + 3, p.w * w);
}

// ---------------------------------------------------------------------------

extern "C" void kernel_launch(void* const* d_in, const int* in_sizes, int n_in,
                              void* d_out, int out_size, void* d_ws, size_t ws_size,
                              hipStream_t stream) {
  const float* x_author = (const float*)d_in[0];   // [100000,128]
  const float* x_field  = (const float*)d_in[1];   // [ 50000,128]
  const float* x_inst   = (const float*)d_in[2];   // [  8000,128]
  const float* x_paper  = (const float*)d_in[3];   // [200000,128]
  const float* weight   = (const float*)d_in[4];   // [7,128,64]
  const float* wself    = (const float*)d_in[5];   // [128,64]
  const float* bias     = (const float*)d_in[6];   // [64]

  const int* e_src[7] = {(const int*)d_in[7],  (const int*)d_in[9],
                         (const int*)d_in[11], (const int*)d_in[13],
                         (const int*)d_in[15], (const int*)d_in[17],
                         (const int*)d_in[19]};
  const int* e_dst[7] = {(const int*)d_in[8],  (const int*)d_in[10],
                         (const int*)d_in[12], (const int*)d_in[14],
                         (const int*)d_in[16], (const int*)d_in[18],
                         (const int*)d_in[20]};

  const int ne[7]   = {1000000, 1000000, 1000000, 800000, 800000, 200000, 200000};
  const float* xsrc[7] = {x_author, x_paper, x_paper, x_paper, x_field, x_author, x_inst};
  const int nsrc[7] = {100000, 200000, 200000, 200000, 50000, 100000, 8000};
  const int ndst[7] = {200000, 100000, 200000, 50000, 200000, 8000, 100000};

  // d_out layout: author[100000,64] | field[50000,64] | inst[8000,64] | paper[200000,64]
  const size_t off_author = 0;
  const size_t off_field  = (size_t)100000 * 64;
  const size_t off_inst   = (size_t)150000 * 64;
  const size_t off_paper  = (size_t)158000 * 64;
  const size_t dst_off[7] = {off_paper, off_author, off_paper, off_field,
                             off_paper, off_inst, off_author};

  float* out  = (float*)d_out;
  float* proj = (float*)d_ws;                        // 200000*64 fp32 = 51.2 MB
  float* deg  = proj + (size_t)200000 * 64;          // 200000 fp32

  // zero accumulators (graph-capturable async memset)
  hipMemsetAsync(d_out, 0, (size_t)358000 * 64 * sizeof(float), stream);

  for (int r = 0; r < 7; ++r) {
    const int gblocks = (nsrc[r] + 127) / 128;
    gemm_proj_kernel<<<gblocks, 256, 0, stream>>>(
        xsrc[r], weight + (size_t)r * K_DIM * N_DIM, proj, nsrc[r]);

    hipMemsetAsync(deg, 0, (size_t)ndst[r] * sizeof(float), stream);
    degree_kernel<<<(ne[r] + 255) / 256, 256, 0, stream>>>(e_dst[r], ne[r], deg);
    invdeg_kernel<<<(ndst[r] + 255) / 256, 256, 0, stream>>>(deg, ndst[r]);

    const long tot = (long)ne[r] * 16;               // 16 threads per edge
    scatter_kernel<<<(unsigned)((tot + 255) / 256), 256, 0, stream>>>(
        proj, e_src[r], e_dst[r], deg, out + dst_off[r], ne[r]);
  }

  finalize_kernel<<<(100000 + 127) / 128, 256, 0, stream>>>(x_author, wself, bias, out + off_author, 100000);
  finalize_kernel<<<( 50000 + 127) / 128, 256, 0, stream>>>(x_field,  wself, bias, out + off_field,   50000);
  finalize_kernel<<<(  8000 + 127) / 128, 256, 0, stream>>>(x_inst,   wself, bias, out + off_inst,     8000);
  finalize_kernel<<<(200000 + 127) / 128, 256, 0, stream>>>(x_paper,  wself, bias, out + off_paper,  200000);
}